// MemoryModule_33346126086839
// MI455X (gfx1250) — compile-verified
//
#include <hip/hip_runtime.h>

// Problem constants from the reference
#define B_ 8
#define L_ 12
#define T_ 36
#define D_ 1024
#define F_ 3
#define C_ 32
#define JR_ 108         // real columns (t*3+f)
#define J_ 112          // padded: 108 real + 1 ones-column (Qsum) + 3 zero pad

typedef float v2f __attribute__((ext_vector_type(2)));
typedef float v8f __attribute__((ext_vector_type(8)));

__device__ __forceinline__ v8f wmma4(v2f a, v2f b, v8f c) {
    // D(16x16 f32) = A(16x4 f32) * B(4x16 f32) + C
    return __builtin_amdgcn_wmma_f32_16x16x4_f32(false, a, false, b, (short)0, c,
                                                 false, false);
}

// ---------------------------------------------------------------------------
// Kernel 1: q[b,c,l,d] = Wq[c,:] . x_local[b,l,d,:] + bq[c]  -> written to out
// ---------------------------------------------------------------------------
__global__ void __launch_bounds__(256) qproj_kernel(const float* __restrict__ xl,
                                                    const float* __restrict__ Wq,
                                                    const float* __restrict__ bq,
                                                    float* __restrict__ out) {
    __shared__ float sW[C_ * F_];
    __shared__ float sb[C_];
    int tid = threadIdx.x;
    if (tid < C_ * F_) sW[tid] = Wq[tid];
    if (tid < C_) sb[tid] = bq[tid];
    __syncthreads();

    int idx = blockIdx.x * 256 + tid;          // flat (b,l,d)
    if (idx >= B_ * L_ * D_) return;
    int b = idx / (L_ * D_);
    int rem = idx - b * (L_ * D_);
    int l = rem / D_;
    int d = rem - l * D_;

    float x0 = xl[idx * 3 + 0];
    float x1 = xl[idx * 3 + 1];
    float x2 = xl[idx * 3 + 2];
#pragma unroll
    for (int c = 0; c < C_; ++c) {
        float qv = sW[c * 3 + 0] * x0 + sW[c * 3 + 1] * x1 + sW[c * 3 + 2] * x2 + sb[c];
        out[((size_t)(b * C_ + c) * L_ + l) * D_ + d] = qv;   // coalesced in d
    }
}

// ---------------------------------------------------------------------------
// Kernel 2: per (b,l):  G[c,j] = sum_d q[c,d] * XH[j,d]   (j = t*3+f, col 108 = ones)
// then mq[c,t] = Wm[c,:].G[c,3t..3t+2] + bm[c]*G[c,108];  att = softmax(relu(mq))
// GEMM M=32, N=112, K=1024 via v_wmma_f32_16x16x4_f32.
// Pad lanes load from a clamped (valid) address and select the constant in,
// so the inner loop has no divergent branches.
// ---------------------------------------------------------------------------
__global__ void __launch_bounds__(256) score_kernel(const float* __restrict__ q,
                                                    const float* __restrict__ xh,
                                                    const float* __restrict__ Wm,
                                                    const float* __restrict__ bm,
                                                    float* __restrict__ att) {
    __shared__ float Gs[C_][J_ + 1];
    __shared__ float mqs[C_][T_];
    __shared__ float sWm[C_ * F_];
    __shared__ float sbm[C_];

    int tid = threadIdx.x;
    int wave = tid >> 5;
    int lane = tid & 31;
    int bl = blockIdx.x;
    int b = bl / L_;
    int l = bl - b * L_;

    if (tid < C_ * F_) sWm[tid] = Wm[tid];
    if (tid >= 128 && tid < 128 + C_) sbm[tid - 128] = bm[tid - 128];

    if (wave < 7) {                                  // 7 N-tiles of 16 cover J_=112
        int nloc = lane & 15;
        int hi = lane >> 4;                          // 0 or 1 (K offset +2)
        int n = wave * 16 + nloc;                    // global column j
        int nc = n < (JR_ - 1) ? n : (JR_ - 1);      // clamp to valid column
        int t = nc / 3;
        int f = nc - t * 3;
        bool real = (n < JR_);
        float padv = (n == JR_) ? 1.0f : 0.0f;       // ones column -> Qsum
        const float* xcol = xh + ((size_t)(bl * T_ + t) * D_) * F_ + f;  // elem(d)=xcol[3d]

        int row0 = lane & 15;                        // A rows (c): 0..15 / 16..31
        const float* qr0 = q + ((size_t)(b * C_ + row0) * L_ + l) * D_;
        const float* qr1 = qr0 + (size_t)16 * L_ * D_;

        v8f acc0 = {0.f, 0.f, 0.f, 0.f, 0.f, 0.f, 0.f, 0.f};
        v8f acc1 = {0.f, 0.f, 0.f, 0.f, 0.f, 0.f, 0.f, 0.f};

        for (int kk = 0; kk < D_; kk += 128) {       // chunked so prefetch is hoisted
            if (kk + 128 < D_) {
                __builtin_prefetch(xcol + (size_t)3 * (kk + 128), 0, 3);
                __builtin_prefetch(xcol + (size_t)3 * (kk + 128) + 128, 0, 3);
            }
#pragma unroll 4
            for (int k = kk; k < kk + 128; k += 4) {
                int k0 = k + 2 * hi;                 // even
                v2f a0 = *(const v2f*)(qr0 + k0);    // global_load_b64
                v2f a1 = *(const v2f*)(qr1 + k0);
                float lx0 = xcol[(size_t)3 * k0];    // unconditional loads
                float lx1 = xcol[(size_t)3 * k0 + 3];
                v2f bv;
                bv.x = real ? lx0 : padv;            // v_cndmask, no branch
                bv.y = real ? lx1 : padv;
                acc0 = wmma4(a0, bv, acc0);
                acc1 = wmma4(a1, bv, acc1);
            }
        }
        // C/D layout: vgpr r -> row r (lanes 0-15) or r+8 (lanes 16-31), col = n
#pragma unroll
        for (int r = 0; r < 8; ++r) {
            Gs[r + hi * 8][n]      = acc0[r];
            Gs[r + hi * 8 + 16][n] = acc1[r];
        }
    }
    __syncthreads();

    if (tid < C_) {
        int c = tid;
        float qsum = Gs[c][JR_];
        float w0 = sWm[c * 3 + 0], w1 = sWm[c * 3 + 1], w2 = sWm[c * 3 + 2];
        float bias = sbm[c] * qsum;
        float mx = 0.0f;                              // relu floor
        for (int t = 0; t < T_; ++t) {
            float s = w0 * Gs[c][3 * t] + w1 * Gs[c][3 * t + 1] + w2 * Gs[c][3 * t + 2] + bias;
            s = s > 0.f ? s : 0.f;                    // relu
            mqs[c][t] = s;
            mx = s > mx ? s : mx;
        }
        float sum = 0.f;
        for (int t = 0; t < T_; ++t) {
            float e = __expf(mqs[c][t] - mx);
            mqs[c][t] = e;
            sum += e;
        }
        float inv = 1.0f / sum;
        float* ap = att + ((size_t)(b * C_ + c) * L_ + l) * T_;
        for (int t = 0; t < T_; ++t) ap[t] = mqs[c][t] * inv;
    }
}

// ---------------------------------------------------------------------------
// Kernel 3: per (b,l, d-chunk of 128):
//   attW[c,j] = att[c, j/3] * Wc[c, j%3]  (j < 108)
//   o[c,d]   = sum_j attW[c,j] * xh[t=j/3, d, f=j%3]
//   out[c,d] += o[c,d] + bc[c]            (out already holds q)
// GEMM M=32, N=128/block (16 per wave), K=108 exactly (27 WMMA steps, no pad).
// B offsets advance by an incremental (t,f) recurrence: no divisions, no guards.
// ---------------------------------------------------------------------------
__global__ void __launch_bounds__(256) out_kernel(const float* __restrict__ xh,
                                                  const float* __restrict__ att,
                                                  const float* __restrict__ Wc,
                                                  const float* __restrict__ bc,
                                                  float* __restrict__ out) {
    __shared__ float sAW[C_][110];                   // stride 110 floats = 440B (8B mult)
    __shared__ float sbc[C_];

    int tid = threadIdx.x;
    int wave = tid >> 5;
    int lane = tid & 31;
    int bl = blockIdx.x;
    int b = bl / L_;
    int l = bl - b * L_;

    for (int i = tid; i < C_ * JR_; i += 256) {
        int c = i / JR_;
        int j = i - c * JR_;
        int t = j / 3;
        int f = j - t * 3;
        sAW[c][j] = att[((size_t)(b * C_ + c) * L_ + l) * T_ + t] * Wc[c * 3 + f];
    }
    if (tid < C_) sbc[tid] = bc[tid];
    __syncthreads();

    int nloc = lane & 15;
    int hi = lane >> 4;
    int d = blockIdx.y * 128 + wave * 16 + nloc;     // output column (d)
    int row0 = lane & 15;
    const float* xd = xh + (size_t)bl * T_ * D_ * F_ + (size_t)3 * d;  // + t*3072 + f

    // Incremental offsets for j0 = k + 2*hi and j1 = j0 + 1 (j = 3t+f -> off = 3072t+f)
    int o0 = hi ? 2 : 0;                             // j0 = 0 -> (0,0); j0 = 2 -> (0,2)
    int f0 = hi ? 2 : 0;
    int o1 = hi ? 3072 : 1;                          // j1 = 1 -> (0,1); j1 = 3 -> (1,0)
    int f1 = hi ? 0 : 1;

    v8f acc0 = {0.f, 0.f, 0.f, 0.f, 0.f, 0.f, 0.f, 0.f};
    v8f acc1 = {0.f, 0.f, 0.f, 0.f, 0.f, 0.f, 0.f, 0.f};

#pragma unroll
    for (int k = 0; k < JR_; k += 4) {               // 27 steps, all j < 108
        int cj = k + 2 * hi;                         // even -> 8B-aligned LDS pair
        v2f a0 = *(const v2f*)(&sAW[row0][cj]);      // ds_load_b64
        v2f a1 = *(const v2f*)(&sAW[row0 + 16][cj]);
        v2f bv;
        bv.x = xd[o0];
        bv.y = xd[o1];
        acc0 = wmma4(a0, bv, acc0);
        acc1 = wmma4(a1, bv, acc1);
        // advance j by 4:  f==2 ? (t+=2,f=0) : (t+=1,f+=1)
        o0 += (f0 == 2) ? (2 * 3072 - 2) : 3073;
        f0 = (f0 == 2) ? 0 : f0 + 1;
        o1 += (f1 == 2) ? (2 * 3072 - 2) : 3073;
        f1 = (f1 == 2) ? 0 : f1 + 1;
    }

#pragma unroll
    for (int r = 0; r < 8; ++r) {
        int c0 = r + hi * 8;
        size_t i0 = ((size_t)(b * C_ + c0) * L_ + l) * D_ + d;
        out[i0] += acc0[r] + sbc[c0];
        int c1 = c0 + 16;
        size_t i1 = ((size_t)(b * C_ + c1) * L_ + l) * D_ + d;
        out[i1] += acc1[r] + sbc[c1];
    }
}

// ---------------------------------------------------------------------------
extern "C" void kernel_launch(void* const* d_in, const int* in_sizes, int n_in,
                              void* d_out, int out_size, void* d_ws, size_t ws_size,
                              hipStream_t stream) {
    const float* xl = (const float*)d_in[0];   // (B,L,D,F)
    const float* xh = (const float*)d_in[1];   // (B,L,T,D,F)
    const float* Wq = (const float*)d_in[2];
    const float* bq = (const float*)d_in[3];
    const float* Wm = (const float*)d_in[4];
    const float* bm = (const float*)d_in[5];
    const float* Wc = (const float*)d_in[6];
    const float* bc = (const float*)d_in[7];
    float* out = (float*)d_out;                // (B,C,L,D) : holds q, then q+o
    float* att = (float*)d_ws;                 // (B,C,L,T) f32 = 442 KB scratch

    // 1) q projection -> out
    hipLaunchKernelGGL(qproj_kernel, dim3((B_ * L_ * D_ + 255) / 256), dim3(256), 0,
                       stream, xl, Wq, bq, out);
    // 2) scores + softmax -> att (one block per (b,l))
    hipLaunchKernelGGL(score_kernel, dim3(B_ * L_), dim3(256), 0, stream,
                       out, xh, Wm, bm, att);
    // 3) attention read + bias + add into out
    hipLaunchKernelGGL(out_kernel, dim3(B_ * L_, D_ / 128), dim3(256), 0, stream,
                       xh, att, Wc, bc, out);
}